// TransducerLoss_35613868818709
// MI455X (gfx1250) — compile-verified
//
#include <hip/hip_runtime.h>
#include <math.h>

#define BB 4
#define TT 200
#define UU 101
#define VV 1024

typedef float f4  __attribute__((ext_vector_type(4)));
typedef float v2f __attribute__((ext_vector_type(2)));
typedef float v8f __attribute__((ext_vector_type(8)));

// Full 32-lane sum using v_wmma_f32_16x16x4_f32 (A = partials striped 16x4,
// B = all-ones 4x16). D[m,n] = s_m + s_{m+16}; summing the 8 D VGPRs gives
// half-wave totals, one xor-16 shuffle completes the reduction on all lanes.
__device__ __forceinline__ float wave_sum32_wmma(float s) {
    v2f a;  a.x = s;    a.y = 0.0f;
    v2f bo; bo.x = 1.0f; bo.y = 1.0f;
    v8f c = {0.f, 0.f, 0.f, 0.f, 0.f, 0.f, 0.f, 0.f};
    v8f d = __builtin_amdgcn_wmma_f32_16x16x4_f32(
        /*neg_a=*/false, a, /*neg_b=*/false, bo,
        /*c_mod=*/(short)0, c, /*reuse_a=*/false, /*reuse_b=*/false);
    float p = d[0] + d[1] + d[2] + d[3] + d[4] + d[5] + d[6] + d[7];
    p += __shfl_xor(p, 16, 32);
    return p;
}

// ---- kernel 0: per-sample lengths -----------------------------------------
__global__ void k_lens(const int* __restrict__ mask, const int* __restrict__ lab_len,
                       int* __restrict__ lens) {
    int b = threadIdx.x;
    if (b < BB) {
        int s = 0;
        for (int t = 0; t < TT; ++t) s += mask[b * TT + t];
        s = s < 1 ? 1 : (s > TT ? TT : s);
        lens[b] = s;
        int u = lab_len[b];
        u = u < 0 ? 0 : (u > UU - 1 ? UU - 1 : u);
        lens[BB + b] = u;
    }
}

// ---- kernel 1: streaming logsumexp + blank/label gather -------------------
// One wave32 per (b,t,u) row of V=1024 floats. Skips rows outside the
// DP-relevant region (t < Tlen, u <= Ulen).
__global__ __launch_bounds__(256) void k_lse(const float* __restrict__ logits,
                                             const int* __restrict__ labels,
                                             const int* __restrict__ lens,
                                             float* __restrict__ blank_lp,
                                             float* __restrict__ label_lp) {
    const int lane = threadIdx.x & 31;
    const int wave = blockIdx.x * (blockDim.x >> 5) + (threadIdx.x >> 5);
    const int NROWS = BB * TT * UU;
    if (wave >= NROWS) return;

    int u  = wave % UU;
    int bt = wave / UU;
    int t  = bt % TT;
    int b  = bt / TT;
    int Tb = lens[b];
    int Ub = lens[BB + b];
    if (t >= Tb || u > Ub) return;   // never feeds the loss

    const f4* rowp = (const f4*)(logits + (size_t)wave * VV);
    f4 x[8];
#pragma unroll
    for (int i = 0; i < 8; ++i)
        x[i] = __builtin_nontemporal_load(&rowp[i * 32 + lane]);  // single-use stream

    int lab = -1;
    if (u < Ub) {
        lab = labels[b * (UU - 1) + u];
        lab = lab < 0 ? 0 : (lab > VV - 1 ? VV - 1 : lab);
    }

    // lane-local max over 32 elements
    float m = -INFINITY;
#pragma unroll
    for (int i = 0; i < 8; ++i)
        m = fmaxf(m, fmaxf(fmaxf(x[i].x, x[i].y), fmaxf(x[i].z, x[i].w)));

    // label logit: owning lane selects it; reduced as a max (others hold -inf)
    float cand = -INFINITY;
    if (lab >= 0 && ((lab >> 2) & 31) == lane) {
        int ii = lab >> 7;
        f4 q = x[0];
#pragma unroll
        for (int i = 1; i < 8; ++i)
            if (ii == i) q = x[i];
        int c = lab & 3;
        cand = (c == 0) ? q.x : (c == 1) ? q.y : (c == 2) ? q.z : q.w;
    }

#pragma unroll
    for (int off = 16; off > 0; off >>= 1) {
        m    = fmaxf(m,    __shfl_xor(m,    off, 32));
        cand = fmaxf(cand, __shfl_xor(cand, off, 32));
    }
    float braw = __shfl(x[0].x, 0, 32);   // element 0 = blank logit

    // lane-local sum of exp(x - m), then WMMA cross-lane reduction
    float s = 0.0f;
#pragma unroll
    for (int i = 0; i < 8; ++i) {
        s += __expf(x[i].x - m);
        s += __expf(x[i].y - m);
        s += __expf(x[i].z - m);
        s += __expf(x[i].w - m);
    }
    float lse = m + __logf(wave_sum32_wmma(s));

    if (lane == 0) {
        blank_lp[wave] = braw - lse;
        if (lab >= 0)
            label_lp[(size_t)(b * TT + t) * (UU - 1) + u] = cand - lse;
    }
}

// ---- kernel 2: anti-diagonal wavefront DP per sample ----------------------
__global__ __launch_bounds__(128) void k_dp(const float* __restrict__ blank_lp,
                                            const float* __restrict__ label_lp,
                                            const int* __restrict__ lens,
                                            float* __restrict__ loss_b) {
    __shared__ float buf[2][UU + 2];
    int b   = blockIdx.x;
    int Tb  = lens[b];
    int Ub  = lens[BB + b];
    int tid = threadIdx.x;
    if (Ub < 1) { if (tid == 0) loss_b[b] = 0.0f; return; }

    const float* bl = blank_lp + (size_t)b * TT * UU;
    const float* lb = label_lp + (size_t)b * TT * (UU - 1);

    int prev = 0;
    int Dlast = (Tb - 1) + Ub;
    for (int d = 0; d <= Dlast; ++d) {
        int cur = prev ^ 1;
        int ulo = d - (Tb - 1); if (ulo < 0) ulo = 0;
        int uhi = d < Ub ? d : Ub;
        int u = ulo + tid;
        if (u <= uhi) {
            int t = d - u;
            float v;
            if (d == 0) {
                v = bl[0];                                  // alpha[0,0] = blank[0,0]
            } else {
                float a1 = (t > 0) ? buf[prev][u]     + bl[t * UU + u]            : -INFINITY;
                float a2 = (u > 0) ? buf[prev][u - 1] + lb[t * (UU - 1) + (u - 1)] : -INFINITY;
                float hi = fmaxf(a1, a2), lo = fminf(a1, a2);
                v = (hi == -INFINITY) ? -INFINITY : hi + log1pf(expf(lo - hi));
            }
            buf[cur][u] = v;
        }
        __syncthreads();
        prev = cur;
    }
    if (tid == 0) loss_b[b] = -buf[prev][Ub];
}

// ---- kernel 3: mean over valid samples ------------------------------------
__global__ void k_fin(const float* __restrict__ loss_b, const int* __restrict__ lens,
                      float* __restrict__ out) {
    if (threadIdx.x == 0 && blockIdx.x == 0) {
        float s = 0.0f; int n = 0;
        for (int b = 0; b < BB; ++b) {
            if (lens[b] > 0 && lens[BB + b] > 0) { s += loss_b[b]; ++n; }
        }
        out[0] = s / (float)(n < 1 ? 1 : n);
    }
}

extern "C" void kernel_launch(void* const* d_in, const int* in_sizes, int n_in,
                              void* d_out, int out_size, void* d_ws, size_t ws_size,
                              hipStream_t stream) {
    const float* logits  = (const float*)d_in[0];
    const int*   labels  = (const int*)d_in[1];
    const int*   lab_len = (const int*)d_in[2];
    const int*   mask    = (const int*)d_in[3];
    float*       out     = (float*)d_out;

    char*  ws       = (char*)d_ws;
    int*   lens     = (int*)ws;                       // 2*B ints
    float* loss_b   = (float*)(ws + 64);              // B floats
    float* blank_lp = (float*)(ws + 128);             // B*T*U floats
    float* label_lp = blank_lp + (size_t)BB * TT * UU; // B*T*(U-1) floats

    k_lens<<<1, 64, 0, stream>>>(mask, lab_len, lens);

    const int nrows  = BB * TT * UU;                  // 80800 waves, 8 per block
    const int blocks = (nrows + 7) / 8;
    k_lse<<<blocks, 256, 0, stream>>>(logits, labels, lens, blank_lp, label_lp);

    k_dp<<<BB, 128, 0, stream>>>(blank_lp, label_lp, lens, loss_b);
    k_fin<<<1, 32, 0, stream>>>(loss_b, lens, out);
}